// WaveBlock_65180423685204
// MI455X (gfx1250) — compile-verified
//
#include <hip/hip_runtime.h>
#include <math.h>

// Sizes fixed by the reference
#define Bb 64
#define Cc 96
#define Hh 64
#define Ww 64
#define HW 4096
#define HID 384

typedef __attribute__((ext_vector_type(2))) float v2f;
typedef __attribute__((ext_vector_type(8))) float v8f;

// ---------------- WMMA helpers (V_WMMA_F32_16X16X4_F32) ----------------
// A 16x4 f32 layout (ISA 7.12.2): lanes 0-15 hold M=lane, VGPR0=K0,VGPR1=K1;
// lanes 16-31 hold M=lane-16, VGPR0=K2,VGPR1=K3.
// B 4x16: row striped across lanes within a VGPR: lanes0-15 K={0,1}, lanes16-31 K={2,3}.
// C/D 16x16: VGPR r: lanes0-15 -> M=r, lanes16-31 -> M=r+8; N=lane&15.

__device__ __forceinline__ v8f wmma4(v2f a, v2f b, v8f c) {
  return __builtin_amdgcn_wmma_f32_16x16x4_f32(
      /*neg_a=*/false, a, /*neg_b=*/false, b,
      /*c_mod=*/(short)0, c, /*reuse_a=*/false, /*reuse_b=*/false);
}

__device__ __forceinline__ v2f ldA(const float* A, int lda, int m0, int k0, int lane) {
  int m = m0 + (lane & 15);
  int k = k0 + ((lane >> 4) << 1);
  const float* p = A + m * lda + k;
  v2f r; r.x = p[0]; r.y = p[1]; return r;
}
// A_frag[m][k] = src[k][m] (transposed source)
__device__ __forceinline__ v2f ldAt(const float* A, int lda, int m0, int k0, int lane) {
  int m = m0 + (lane & 15);
  int k = k0 + ((lane >> 4) << 1);
  v2f r; r.x = A[k * lda + m]; r.y = A[(k + 1) * lda + m]; return r;
}
__device__ __forceinline__ v2f ldB(const float* Bm, int ldb, int k0, int n0, int lane) {
  int n = n0 + (lane & 15);
  int k = k0 + ((lane >> 4) << 1);
  v2f r; r.x = Bm[k * ldb + n]; r.y = Bm[(k + 1) * ldb + n]; return r;
}
// B_frag[k][n] = src[n][k] (transposed source)
__device__ __forceinline__ v2f ldBt(const float* Bm, int ldb, int k0, int n0, int lane) {
  int n = n0 + (lane & 15);
  int k = k0 + ((lane >> 4) << 1);
  const float* p = Bm + n * ldb + k;
  v2f r; r.x = p[0]; r.y = p[1]; return r;
}
__device__ __forceinline__ void stTile(float* Dst, int ldd, int m0, int n0, int lane, v8f acc) {
  int col = n0 + (lane & 15);
  int rb = m0 + ((lane >> 4) << 3);
#pragma unroll
  for (int r = 0; r < 8; ++r) Dst[(rb + r) * ldd + col] = acc[r];
}

// ---------------- k0a: orthonormal DCT-II matrix D[64][64] ----------------
__global__ void k0_dct(float* __restrict__ D) {
  int n = threadIdx.x, k = blockIdx.x;
  float v = cosf((float)k * ((float)n + 0.5f) * (3.14159265358979f / 64.f)) * 0.17677669529f;
  if (k == 0) v *= 0.70710678119f;
  D[k * 64 + n] = v;
}

// ---------------- k0b: phase field PH[c][h][w] ----------------
// t = relu(tok_w @ fe + tok_b); PH = cos(c*t) + sin(c*t)/c * (1 + 0.5*alpha)
__global__ void k0_phase(const float* __restrict__ fe, const float* __restrict__ tw,
                         const float* __restrict__ tb, const float* __restrict__ cs,
                         const float* __restrict__ al, float* __restrict__ PH) {
  int tid = blockIdx.x * 256 + threadIdx.x;  // 96*4096 threads
  int d = tid >> 12;
  int hw = tid & 4095;
  const float* fp = fe + hw * Cc;
  const float* wp = tw + d * Cc;
  float t = tb[d];
  for (int c = 0; c < Cc; ++c) t += fp[c] * wp[c];
  t = fmaxf(t, 0.f);
  float c0 = cs[0], a0 = al[0];
  PH[tid] = cosf(c0 * t) + sinf(c0 * t) / c0 * (1.f + 0.5f * a0);
}

// ---------------- k1: depthwise 3x3 conv + lin GEMM (192x96 @ 96x64px) ----
__global__ void k1_dwlin(const float* __restrict__ x, const float* __restrict__ dww,
                         const float* __restrict__ dwb, const float* __restrict__ linw,
                         const float* __restrict__ linb, float* __restrict__ XW,
                         float* __restrict__ SZ) {
  __shared__ float Sh[Cc * 64];
  int blk = blockIdx.x;                 // 4096 = B * (HW/64)
  int b = blk >> 6;
  int p0 = (blk & 63) << 6;
  int tid = threadIdx.x;
  // Stage A: depthwise conv tile into LDS
  for (int i = tid; i < Cc * 64; i += 256) {
    int c = i >> 6, j = i & 63;
    int p = p0 + j;
    int yy = p >> 6, xx = p & 63;
    const float* xb = x + (((size_t)b * Cc + c) << 12);
    const float* wq = dww + c * 9;
    float s = dwb[c];
#pragma unroll
    for (int dy = -1; dy <= 1; ++dy) {
      int yv = yy + dy;
      if (yv < 0 || yv >= 64) continue;
#pragma unroll
      for (int dx = -1; dx <= 1; ++dx) {
        int xv = xx + dx;
        if (xv < 0 || xv >= 64) continue;
        s += xb[(yv << 6) + xv] * wq[(dy + 1) * 3 + (dx + 1)];
      }
    }
    Sh[i] = s;
  }
  __syncthreads();
  // Stage B: y = lin_w[192x96] @ Sh[96x64]; 48 tiles, 8 waves x 6
  int lane = tid & 31, wid = tid >> 5;
  for (int t6 = 0; t6 < 6; ++t6) {
    int t = wid * 6 + t6;
    int m0 = (t >> 2) << 4, n0 = (t & 3) << 4;
    v8f acc = {};
    for (int k0 = 0; k0 < Cc; k0 += 4)
      acc = wmma4(ldA(linw, Cc, m0, k0, lane), ldB(Sh, 64, k0, n0, lane), acc);
    int col = n0 + (lane & 15);
    int rb = m0 + ((lane >> 4) << 3);
#pragma unroll
    for (int r = 0; r < 8; ++r) {
      int d = rb + r;
      float v = acc[r] + linb[d];
      if (d < Cc) {
        XW[(((size_t)b * Cc + d) << 12) + p0 + col] = v;
      } else {
        float sg = v / (1.f + expf(-v));  // silu(z)
        SZ[(((size_t)b * Cc + (d - Cc)) << 12) + p0 + col] = sg;
      }
    }
  }
}

// ---------------- k2: per (b,c) plane: DCT -> phase -> IDCT, in-place -----
__global__ void k2_dct(float* __restrict__ XW, const float* __restrict__ Dg,
                       const float* __restrict__ PH) {
  __shared__ float Dl[4096];
  __shared__ float Sa[4096];
  __shared__ float Sb[4096];
  int plane = blockIdx.x;  // B*C planes
  int c = plane % Cc;
  float* xp = XW + ((size_t)plane << 12);
  int tid = threadIdx.x;
  for (int i = tid; i < 4096; i += 256) { Dl[i] = Dg[i]; Sa[i] = xp[i]; }
  __syncthreads();
  int lane = tid & 31, wid = tid >> 5;
  const float* ph = PH + ((size_t)c << 12);
  // Stage 1: Sb = D @ Sa
  for (int t = wid * 2; t < wid * 2 + 2; ++t) {
    int m0 = (t >> 2) << 4, n0 = (t & 3) << 4;
    v8f acc = {};
    for (int k0 = 0; k0 < 64; k0 += 4)
      acc = wmma4(ldA(Dl, 64, m0, k0, lane), ldB(Sa, 64, k0, n0, lane), acc);
    stTile(Sb, 64, m0, n0, lane, acc);
  }
  __syncthreads();
  // Stage 2: Sa = (Sb @ D^T) * PH
  for (int t = wid * 2; t < wid * 2 + 2; ++t) {
    int m0 = (t >> 2) << 4, n0 = (t & 3) << 4;
    v8f acc = {};
    for (int k0 = 0; k0 < 64; k0 += 4)
      acc = wmma4(ldA(Sb, 64, m0, k0, lane), ldBt(Dl, 64, k0, n0, lane), acc);
    int col = n0 + (lane & 15);
    int rb = m0 + ((lane >> 4) << 3);
#pragma unroll
    for (int r = 0; r < 8; ++r) acc[r] *= ph[((rb + r) << 6) + col];
    stTile(Sa, 64, m0, n0, lane, acc);
  }
  __syncthreads();
  // Stage 3: Sb = D^T @ Sa
  for (int t = wid * 2; t < wid * 2 + 2; ++t) {
    int m0 = (t >> 2) << 4, n0 = (t & 3) << 4;
    v8f acc = {};
    for (int k0 = 0; k0 < 64; k0 += 4)
      acc = wmma4(ldAt(Dl, 64, m0, k0, lane), ldB(Sa, 64, k0, n0, lane), acc);
    stTile(Sb, 64, m0, n0, lane, acc);
  }
  __syncthreads();
  // Stage 4: xf = Sb @ D  -> global (in-place)
  for (int t = wid * 2; t < wid * 2 + 2; ++t) {
    int m0 = (t >> 2) << 4, n0 = (t & 3) << 4;
    v8f acc = {};
    for (int k0 = 0; k0 < 64; k0 += 4)
      acc = wmma4(ldA(Sb, 64, m0, k0, lane), ldB(Dl, 64, k0, n0, lane), acc);
    int col = n0 + (lane & 15);
    int rb = m0 + ((lane >> 4) << 3);
#pragma unroll
    for (int r = 0; r < 8; ++r) xp[((rb + r) << 6) + col] = acc[r];
  }
}

// ---------------- k3: out_norm LN + gate + olin GEMM + residual + LN(n1) --
__global__ void k3_gate(const float* __restrict__ x, const float* __restrict__ XW,
                        const float* __restrict__ SZ, const float* __restrict__ og,
                        const float* __restrict__ ob, const float* __restrict__ olw,
                        const float* __restrict__ olb, const float* __restrict__ n1g,
                        const float* __restrict__ n1b, float* __restrict__ out) {
  __shared__ float Sx[Cc * 64];
  __shared__ float So[Cc * 64];
  int blk = blockIdx.x;
  int b = blk >> 6;
  int p0 = (blk & 63) << 6;
  int tid = threadIdx.x;
  size_t base = ((size_t)b * Cc) << 12;
  for (int i = tid; i < Cc * 64; i += 256) {
    int c = i >> 6, j = i & 63;
    Sx[i] = XW[base + ((size_t)c << 12) + p0 + j];
  }
  __syncthreads();
  if (tid < 64) {  // channel-LN (eps 1e-5) then * silu(z)
    int j = tid;
    float mu = 0.f;
    for (int c = 0; c < Cc; ++c) mu += Sx[(c << 6) + j];
    mu *= (1.f / Cc);
    float var = 0.f;
    for (int c = 0; c < Cc; ++c) { float d = Sx[(c << 6) + j] - mu; var += d * d; }
    var *= (1.f / Cc);
    float rs = rsqrtf(var + 1e-5f);
    for (int c = 0; c < Cc; ++c) {
      float xn = (Sx[(c << 6) + j] - mu) * rs * og[c] + ob[c];
      float g = SZ[base + ((size_t)c << 12) + p0 + j];
      Sx[(c << 6) + j] = xn * g;
    }
  }
  __syncthreads();
  int lane = tid & 31, wid = tid >> 5;
  // op = olin_w[96x96] @ Sx[96x64]; 24 tiles, 8 waves x 3
  for (int t3 = 0; t3 < 3; ++t3) {
    int t = wid * 3 + t3;
    int m0 = (t >> 2) << 4, n0 = (t & 3) << 4;
    v8f acc = {};
    for (int k0 = 0; k0 < Cc; k0 += 4)
      acc = wmma4(ldA(olw, Cc, m0, k0, lane), ldB(Sx, 64, k0, n0, lane), acc);
    int col = n0 + (lane & 15);
    int rb = m0 + ((lane >> 4) << 3);
#pragma unroll
    for (int r = 0; r < 8; ++r) So[((rb + r) << 6) + col] = acc[r] + olb[rb + r];
  }
  __syncthreads();
  if (tid < 64) {  // x1 = x + LN(op, eps 1e-6)
    int j = tid;
    float mu = 0.f;
    for (int c = 0; c < Cc; ++c) mu += So[(c << 6) + j];
    mu *= (1.f / Cc);
    float var = 0.f;
    for (int c = 0; c < Cc; ++c) { float d = So[(c << 6) + j] - mu; var += d * d; }
    var *= (1.f / Cc);
    float rs = rsqrtf(var + 1e-6f);
    for (int c = 0; c < Cc; ++c) {
      size_t gi = base + ((size_t)c << 12) + p0 + j;
      out[gi] = x[gi] + (So[(c << 6) + j] - mu) * rs * n1g[c] + n1b[c];
    }
  }
}

// ---------------- k4: fused MLP + LN(n2) + residual, in-place on d_out ----
__global__ void k4_mlp(const float* __restrict__ f1w, const float* __restrict__ f1b,
                       const float* __restrict__ f2w, const float* __restrict__ f2b,
                       const float* __restrict__ n2g, const float* __restrict__ n2b,
                       float* __restrict__ out) {
  __shared__ float Sx[Cc * 32];    // x1 tile
  __shared__ float Sh[HID * 32];   // hidden; reused for m
  int blk = blockIdx.x;            // 8192 = B * (HW/32)
  int b = blk >> 7;
  int p0 = (blk & 127) << 5;
  int tid = threadIdx.x;
  size_t base = ((size_t)b * Cc) << 12;
  for (int i = tid; i < Cc * 32; i += 256) {
    int c = i >> 5, j = i & 31;
    Sx[i] = out[base + ((size_t)c << 12) + p0 + j];
  }
  __syncthreads();
  int lane = tid & 31, wid = tid >> 5;
  // GEMM1: h = gelu(fc1_w[384x96] @ Sx[96x32] + b); 48 tiles, 8 waves x 6
  for (int t6 = 0; t6 < 6; ++t6) {
    int t = wid * 6 + t6;
    int m0 = (t >> 1) << 4, n0 = (t & 1) << 4;
    v8f acc = {};
    for (int k0 = 0; k0 < Cc; k0 += 4)
      acc = wmma4(ldA(f1w, Cc, m0, k0, lane), ldB(Sx, 32, k0, n0, lane), acc);
    int col = n0 + (lane & 15);
    int rb = m0 + ((lane >> 4) << 3);
#pragma unroll
    for (int r = 0; r < 8; ++r) {
      int d = rb + r;
      float v = acc[r] + f1b[d];
      v = 0.5f * v * (1.f + erff(v * 0.70710678f));  // exact GELU
      Sh[(d << 5) + col] = v;
    }
  }
  __syncthreads();
  // GEMM2: m = fc2_w[96x384] @ Sh[384x32]; 12 tiles
  v8f a0 = {}, a1 = {};
  int tl[2]; int nt = 0;
  for (int t = wid; t < 12; t += 8) tl[nt++] = t;
  for (int q = 0; q < nt; ++q) {
    int t = tl[q];
    int m0 = (t >> 1) << 4, n0 = (t & 1) << 4;
    v8f acc = {};
    for (int k0 = 0; k0 < HID; k0 += 4)
      acc = wmma4(ldA(f2w, HID, m0, k0, lane), ldB(Sh, 32, k0, n0, lane), acc);
    if (q == 0) a0 = acc; else a1 = acc;
  }
  __syncthreads();  // Sh no longer needed as B operand; reuse for m
  for (int q = 0; q < nt; ++q) {
    int t = tl[q];
    int m0 = (t >> 1) << 4, n0 = (t & 1) << 4;
    v8f acc = q ? a1 : a0;
    int col = n0 + (lane & 15);
    int rb = m0 + ((lane >> 4) << 3);
#pragma unroll
    for (int r = 0; r < 8; ++r) Sh[((rb + r) << 5) + col] = acc[r] + f2b[rb + r];
  }
  __syncthreads();
  if (tid < 32) {  // out = x1 + LN(m, eps 1e-6)
    int j = tid;
    float mu = 0.f;
    for (int c = 0; c < Cc; ++c) mu += Sh[(c << 5) + j];
    mu *= (1.f / Cc);
    float var = 0.f;
    for (int c = 0; c < Cc; ++c) { float d = Sh[(c << 5) + j] - mu; var += d * d; }
    var *= (1.f / Cc);
    float rs = rsqrtf(var + 1e-6f);
    for (int c = 0; c < Cc; ++c)
      out[base + ((size_t)c << 12) + p0 + j] =
          Sx[(c << 5) + j] + (Sh[(c << 5) + j] - mu) * rs * n2g[c] + n2b[c];
  }
}

extern "C" void kernel_launch(void* const* d_in, const int* in_sizes, int n_in,
                              void* d_out, int out_size, void* d_ws, size_t ws_size,
                              hipStream_t stream) {
  const float* x    = (const float*)d_in[0];
  const float* fe   = (const float*)d_in[1];
  const float* dww  = (const float*)d_in[2];
  const float* dwb  = (const float*)d_in[3];
  const float* linw = (const float*)d_in[4];
  const float* linb = (const float*)d_in[5];
  const float* tokw = (const float*)d_in[6];
  const float* tokb = (const float*)d_in[7];
  const float* cs   = (const float*)d_in[8];
  const float* al   = (const float*)d_in[9];
  const float* og   = (const float*)d_in[10];
  const float* ob   = (const float*)d_in[11];
  const float* olw  = (const float*)d_in[12];
  const float* olb  = (const float*)d_in[13];
  const float* n1g  = (const float*)d_in[14];
  const float* n1b  = (const float*)d_in[15];
  const float* n2g  = (const float*)d_in[16];
  const float* n2b  = (const float*)d_in[17];
  const float* f1w  = (const float*)d_in[18];
  const float* f1b  = (const float*)d_in[19];
  const float* f2w  = (const float*)d_in[20];
  const float* f2b  = (const float*)d_in[21];
  float* out = (float*)d_out;

  float* ws = (float*)d_ws;
  float* D  = ws;                         // 4096 floats
  float* PH = ws + 4096;                  // 96*4096 floats
  float* XW = PH + (size_t)Cc * HW;       // B*C*HW floats (xw, then xf in-place)
  float* SZ = XW + (size_t)Bb * Cc * HW;  // B*C*HW floats (silu(z))

  k0_dct<<<64, 64, 0, stream>>>(D);
  k0_phase<<<(Cc * HW) / 256, 256, 0, stream>>>(fe, tokw, tokb, cs, al, PH);
  k1_dwlin<<<Bb * (HW / 64), 256, 0, stream>>>(x, dww, dwb, linw, linb, XW, SZ);
  k2_dct<<<Bb * Cc, 256, 0, stream>>>(XW, D, PH);
  k3_gate<<<Bb * (HW / 64), 256, 0, stream>>>(x, XW, SZ, og, ob, olw, olb, n1g, n1b, out);
  k4_mlp<<<Bb * (HW / 32), 256, 0, stream>>>(f1w, f1b, f2w, f2b, n2g, n2b, out);
}